// GraphAttentionLayer_34110630265699
// MI455X (gfx1250) — compile-verified
//
#include <hip/hip_runtime.h>

// GraphAttentionLayer, algebraically reduced to two bandwidth-bound GEMV
// passes over adj (the only large operand: 32768x1024 f32 = 128 MB), plus a
// WMMA (v_wmma_f32_16x16x4_f32) tail projection p @ W_1.
//
// N=32768, F_IN=1024, F_OUT=512.
// Traffic: adj twice (256 MB) + ~8 MB small ops -> ~11 us at 23.3 TB/s.

#define NROWS 32768
#define FIN   1024
#define FOUT  512
#define ALPHA 0.1f

typedef __attribute__((ext_vector_type(2))) float v2f;
typedef __attribute__((ext_vector_type(8))) float v8f;

__device__ __forceinline__ float waveRedSum(float v) {
#pragma unroll
  for (int m = 16; m >= 1; m >>= 1) v += __shfl_xor(v, m, 32);
  return v;
}
__device__ __forceinline__ float waveRedMax(float v) {
#pragma unroll
  for (int m = 16; m >= 1; m >>= 1) v = fmaxf(v, __shfl_xor(v, m, 32));
  return v;
}

// ---- K1: u[k] = sum_j W[k,j]*a2[j], v[k] = sum_j W[k,j]*a1[j].
// One wave per k (1024 waves = 128 blocks x 8 waves). Coalesced row reads.
__global__ void k_uv(const float* __restrict__ W, const float* __restrict__ a,
                     float* __restrict__ u, float* __restrict__ v) {
  const int wv   = threadIdx.x >> 5;
  const int lane = threadIdx.x & 31;
  const int k    = blockIdx.x * 8 + wv;          // 0..1023
  const float* wrow = W + (size_t)k * FOUT;
  float su = 0.f, sv = 0.f;
#pragma unroll 4
  for (int j = lane; j < FOUT; j += 32) {
    float w = wrow[j];
    su = fmaf(w, a[FOUT + j], su);               // a2
    sv = fmaf(w, a[j],        sv);               // a1
  }
  su = waveRedSum(su);
  sv = waveRedSum(sv);
  if (lane == 0) { u[k] = su; v[k] = sv; }
}

// ---- K2: c0 = h . v  (single block).
__global__ void k_c0(const float* __restrict__ h, const float* __restrict__ v,
                     float* __restrict__ c0) {
  __shared__ float sm[8];
  const int tid = threadIdx.x, wv = tid >> 5, lane = tid & 31;
  const float4 hv = ((const float4*)h)[tid];     // 256 threads * 4 = 1024
  const float4 vv = ((const float4*)v)[tid];
  float part = hv.x * vv.x + hv.y * vv.y + hv.z * vv.z + hv.w * vv.w;
  part = waveRedSum(part);
  if (lane == 0) sm[wv] = part;
  __syncthreads();
  if (tid == 0) {
    float t = 0.f;
#pragma unroll
    for (int i = 0; i < 8; ++i) t += sm[i];
    c0[0] = t;
  }
}

// ---- K3: scores[i] = leaky_relu(c0 + adj[i,:] . u). Streams adj once.
// 1024 blocks x 256 threads; 32 rows/block; float4 coalesced loads.
__global__ void k_scores(const float* __restrict__ adj, const float* __restrict__ u,
                         const float* __restrict__ c0, float* __restrict__ scores) {
  __shared__ float sm[8];
  const int tid = threadIdx.x, wv = tid >> 5, lane = tid & 31;
  const float4 u4  = ((const float4*)u)[tid];    // per-thread column slice, reused
  const float  c0v = c0[0];
  const int r0 = blockIdx.x * 32;
  const float4* adj4 = (const float4*)adj;
  for (int i = r0; i < r0 + 32; ++i) {
    const float4 av = adj4[(size_t)i * (FIN / 4) + tid];
    float part = av.x * u4.x + av.y * u4.y + av.z * u4.z + av.w * u4.w;
    part = waveRedSum(part);
    if (lane == 0) sm[wv] = part;
    __syncthreads();
    if (tid == 0) {
      float s = c0v;
#pragma unroll
      for (int q = 0; q < 8; ++q) s += sm[q];
      scores[i] = (s > 0.f) ? s : ALPHA * s;
    }
    __syncthreads();
  }
}

// ---- K4: softmax over 32768 scores; overwrite scores with attention;
// attention2 = att > mean(att) = att > 1/N. Single block of 1024 threads.
__global__ void k_softmax(float* __restrict__ scores, int* __restrict__ att2) {
  __shared__ float sm[32];
  const int tid = threadIdx.x, wv = tid >> 5, lane = tid & 31;
  // block max
  float lm = -3.402823466e38f;
#pragma unroll
  for (int r = 0; r < NROWS / 1024; ++r) lm = fmaxf(lm, scores[tid + 1024 * r]);
  lm = waveRedMax(lm);
  if (lane == 0) sm[wv] = lm;
  __syncthreads();
  if (wv == 0) { float x = waveRedMax(sm[lane]); if (lane == 0) sm[0] = x; }
  __syncthreads();
  const float mx = sm[0];
  __syncthreads();
  // block sum of exp
  float ls = 0.f;
#pragma unroll
  for (int r = 0; r < NROWS / 1024; ++r) ls += __expf(scores[tid + 1024 * r] - mx);
  ls = waveRedSum(ls);
  if (lane == 0) sm[wv] = ls;
  __syncthreads();
  if (wv == 0) { float x = waveRedSum(sm[lane]); if (lane == 0) sm[0] = x; }
  __syncthreads();
  const float invZ = 1.0f / sm[0];
  const float thr  = 1.0f / (float)NROWS;        // mean(softmax) == 1/N
#pragma unroll
  for (int r = 0; r < NROWS / 1024; ++r) {
    const int idx = tid + 1024 * r;
    float att = __expf(scores[idx] - mx) * invZ;
    scores[idx] = att;                           // reuse buffer as attention
    att2[idx]   = (att > thr) ? 1 : 0;
  }
}

// ---- K5: per-block partial of p[k] = sum_i att[i]*adj[i,k]. Streams adj again.
// 512 blocks x 256 threads; 64 rows/block; thread t owns cols 4t..4t+3.
__global__ void k_wsum(const float* __restrict__ adj, const float* __restrict__ att,
                       float* __restrict__ partials) {
  __shared__ float satt[64];
  const int tid = threadIdx.x;
  const int r0  = blockIdx.x * 64;
  if (tid < 64) satt[tid] = att[r0 + tid];
  __syncthreads();
  const float4* adj4 = (const float4*)adj;
  float4 acc = make_float4(0.f, 0.f, 0.f, 0.f);
#pragma unroll 4
  for (int r = 0; r < 64; ++r) {
    const float  w  = satt[r];
    const float4 av = adj4[(size_t)(r0 + r) * (FIN / 4) + tid];
    acc.x = fmaf(w, av.x, acc.x);
    acc.y = fmaf(w, av.y, acc.y);
    acc.z = fmaf(w, av.z, acc.z);
    acc.w = fmaf(w, av.w, acc.w);
  }
  ((float4*)(partials + (size_t)blockIdx.x * FIN))[tid] = acc;
}

// ---- K6: p[c] = sum_b partials[b][c]. 4 blocks x 256 threads (fixed order).
__global__ void k_reduce_p(const float* __restrict__ partials, float* __restrict__ p) {
  const int c = blockIdx.x * 256 + threadIdx.x;
  float s = 0.f;
  for (int b = 0; b < 512; ++b) s += partials[(size_t)b * FIN + c];
  p[c] = s;
}

// ---- K7: out[n] = sum_k p[k]*W1[k,n] via V_WMMA_F32_16X16X4_F32.
// A (16x4 f32): row M=0 carries p[k0..k0+3] (lane0: K0,K1 ; lane16: K2,K3),
// other rows zero. Branchless: every lane loads p[k0 + 2*(lane>>4) + {0,1}]
// (two broadcast addresses) and multiplies by a hoisted 0/1 mask, so the WMMA
// loop has no EXEC manipulation. B (4x16 f32): VGPR0 = {K0 lanes0-15,
// K2 lanes16-31}, VGPR1 = {K1, K3}, N = lane&15. D row M=0 = acc[0] on
// lanes 0-15. 8 waves x 4 tiles of 16 cols cover all 512 outputs.
__global__ void k_out_wmma(const float* __restrict__ p, const float* __restrict__ W1,
                           float* __restrict__ out) {
  const int wv = threadIdx.x >> 5, lane = threadIdx.x & 31;
  const float msk  = ((lane & 15) == 0) ? 1.0f : 0.0f;  // lanes 0 and 16 only
  const int   khalf = (lane >> 4) << 1;                 // 0 or 2
  for (int t = 0; t < 4; ++t) {
    const int n0 = wv * 64 + t * 16;
    v8f acc = (v8f)(0.f);
    for (int k0 = 0; k0 < FIN; k0 += 4) {
      const int kk = k0 + khalf;
      v2f A; A[0] = msk * p[kk]; A[1] = msk * p[kk + 1];
      const int nn = n0 + (lane & 15);
      v2f B; B[0] = W1[(size_t)kk * FOUT + nn]; B[1] = W1[(size_t)(kk + 1) * FOUT + nn];
      // 8 args: (neg_a, A, neg_b, B, c_mod, C, reuse_a, reuse_b)
      acc = __builtin_amdgcn_wmma_f32_16x16x4_f32(false, A, false, B, (short)0, acc,
                                                  false, false);
    }
    if (lane < 16) out[n0 + lane] = acc[0];
  }
}

extern "C" void kernel_launch(void* const* d_in, const int* in_sizes, int n_in,
                              void* d_out, int out_size, void* d_ws, size_t ws_size,
                              hipStream_t stream) {
  (void)in_sizes; (void)n_in; (void)out_size; (void)ws_size;
  const float* h   = (const float*)d_in[0];   // 1024
  const float* adj = (const float*)d_in[1];   // 32768*1024
  const float* W   = (const float*)d_in[2];   // 1024*512
  const float* a   = (const float*)d_in[3];   // 1024
  const float* W1  = (const float*)d_in[4];   // 1024*512

  float* outp = (float*)d_out;                // out: 512 floats
  int*   att2 = (int*)d_out + FOUT;           // attention2: 32768 int32

  float* ws       = (float*)d_ws;
  float* u        = ws;                       // 1024
  float* v        = ws + 1024;                // 1024
  float* c0       = ws + 2048;                // 1
  float* scores   = ws + 2560;                // 32768 (becomes attention)
  float* p        = ws + 2560 + NROWS;        // 1024
  float* partials = ws + 2560 + NROWS + 1024; // 512*1024 (~2 MB)

  k_uv      <<<128,  256, 0, stream>>>(W, a, u, v);
  k_c0      <<<1,    256, 0, stream>>>(h, v, c0);
  k_scores  <<<1024, 256, 0, stream>>>(adj, u, c0, scores);
  k_softmax <<<1,   1024, 0, stream>>>(scores, att2);
  k_wsum    <<<512,  256, 0, stream>>>(adj, scores, partials);
  k_reduce_p<<<4,    256, 0, stream>>>(partials, p);
  k_out_wmma<<<1,    256, 0, stream>>>(p, W1, outp);
}